// SeqLabel_51170240365077
// MI455X (gfx1250) — compile-verified
//
#include <hip/hip_runtime.h>
#include <stdint.h>

// ---------------------------------------------------------------------------
// CRF forward (log-partition) via f32 WMMA + Tensor Data Mover on gfx1250.
//
//   NUM_LABELS=49, LABEL_SIZE=51 (START=49, END=50), B=1024, S=512, NEG=-100
//
// Reformulation: E[i][j] = exp(T[i][j]) (constant), then per step t:
//   v[j,b]   = exp(alpha[j,b] - m[b])          (m = per-column max, stabilizer)
//   s[i,b]   = sum_j E[i][j] * v[j,b]          (64x64 @ 64x16 f32 GEMM -> WMMA)
//   alpha'   = lp[i,b,t] + m[b] + log(s[i,b])  (frozen columns keep old alpha)
// Pad L=51 -> 64; pad rows of E are 0 so padded alpha stays -inf (-1e30).
// K is processed in 4-wide chunks; chunks 13..15 (j=52..63) are all padding
// (zero A-fragments), so only 13 chunks are issued.
// ---------------------------------------------------------------------------

#define NL    49
#define L     51
#define LP    64
#define SSEQ  512
#define BB    1024
#define START 49
#define ENDL  50
#define KCH   13            // ceil(52/4): K chunks with any real data

typedef float v2f  __attribute__((ext_vector_type(2)));
typedef float v8f  __attribute__((ext_vector_type(8)));
typedef unsigned int u32x4 __attribute__((ext_vector_type(4)));
typedef int   i32x8 __attribute__((ext_vector_type(8)));
typedef int   i32x4 __attribute__((ext_vector_type(4)));

// ---------------------------------------------------------------------------
// TDM: load a [16 rows (batch) x 49 floats] tile into LDS.
//   row stride in memory = S*49 elements (element size 4B).
// D# group0: count=1 | lds_addr | global_addr(57b) | type=2 in bits 127:126
// D# group1: data_size=2(4B), tensor_dim0=49, tensor_dim1=16,
//            tile_dim0=49, tile_dim1=16, tensor_dim0_stride=S*49=25088
// This toolchain's builtin takes 6 args (extra i32x8 group before cpol).
// ---------------------------------------------------------------------------
static __device__ __forceinline__ void tdm_load_tile(const float* gptr, unsigned lds_off)
{
    unsigned long long ga = (unsigned long long)(uintptr_t)gptr;
    u32x4 g0;
    g0[0] = 1u;                                   // count=1, user descriptor
    g0[1] = lds_off;                              // LDS byte address
    g0[2] = (unsigned)(ga & 0xffffffffu);         // global addr [31:0]
    g0[3] = (unsigned)((ga >> 32) & 0x01ffffffu)  // global addr [56:32]
          | 0x80000000u;                          // type = 2 ("image")
    i32x8 g1;
    g1[0] = 0x00020000;        // workgroup_mask=0, data_size=2 (4 bytes)
    g1[1] = (49 << 16);        // tensor_dim0[15:0] = 49
    g1[2] = (16 << 16);        // tensor_dim0 hi = 0 | tensor_dim1[15:0] = 16
    g1[3] = (49 << 16);        // tensor_dim1 hi = 0 | tile_dim0 = 49
    g1[4] = 16;                // tile_dim1 = 16, tile_dim2 = 0
    g1[5] = 25088;             // tensor_dim0_stride = S*49 (elements)
    g1[6] = 0;                 // stride0 hi, tensor_dim1_stride lo
    g1[7] = 0;
    i32x4 z4 = {0, 0, 0, 0};                      // groups 2/3 unused (2D tile)
    i32x8 z8 = {0, 0, 0, 0, 0, 0, 0, 0};          // extra group (6-arg form)
    __builtin_amdgcn_tensor_load_to_lds(g0, g1, z4, z4, z8, 0);
}

// ---------------------------------------------------------------------------
// Gold path score: one wave per batch element.
// ---------------------------------------------------------------------------
__global__ void __launch_bounds__(256)
crf_gold_kernel(const float* __restrict__ logits, const float* __restrict__ T,
                const int* __restrict__ labels, const int* __restrict__ lens,
                float* __restrict__ gold)
{
    __shared__ float Ts[L * L];
    for (int idx = threadIdx.x; idx < L * L; idx += 256) Ts[idx] = T[idx];
    __syncthreads();

    const int wid  = threadIdx.x >> 5;
    const int lane = threadIdx.x & 31;
    const int b    = blockIdx.x * 8 + wid;

    const int len = lens[b];
    const int*   lb = labels + (size_t)b * SSEQ;
    const float* lg = logits + (size_t)b * SSEQ * NL;

    float partial = 0.0f;
    for (int t = lane; t < len; t += 32) {
        int cur  = lb[t];
        int prev = (t == 0) ? START : lb[t - 1];
        partial += lg[t * NL + cur] + Ts[cur * L + prev];
    }
    #pragma unroll
    for (int off = 16; off > 0; off >>= 1)
        partial += __shfl_xor(partial, off, 32);

    if (lane == 0) {
        int lastlab = lb[len - 1];                 // len >= 1 guaranteed
        gold[b] = partial + Ts[ENDL * L + lastlab];
    }
}

// ---------------------------------------------------------------------------
// Main forward kernel. One workgroup (4 waves) per 16-column batch tile.
// Wave w computes output rows [16w, 16w+16) via V_WMMA_F32_16X16X4_F32;
// its 13 A-fragments of E (rows x K-chunks) are held in registers for the
// entire scan. v is exchanged through LDS in B-fragment layout.
// ---------------------------------------------------------------------------
__global__ void __launch_bounds__(128)
crf_fwd_kernel(const float* __restrict__ logits, const float* __restrict__ T,
               const int* __restrict__ lens, const float* __restrict__ gold,
               float* __restrict__ out)
{
    // B-fragment friendly v buffer: rows paired; pair p, col b -> 2 floats.
    __shared__ float vbuf[LP * 16];            // 4 KB
    __shared__ float lbuf[2][16 * NL];         // TDM double buffer (2x3136 B)
    __shared__ float pm[128];                  // per-(wave,half,col) max
    __shared__ float psum[128];                // per-(wave,half,col) sum
    __shared__ float tEnd[LP];                 // T[END][i] padded

    const int tid  = threadIdx.x;
    const int wid  = tid >> 5;
    const int lane = tid & 31;
    const int col  = lane & 15;                // batch column in tile
    const int half = lane >> 4;                // wave half
    const int b0   = blockIdx.x * 16;

    if (tid < LP) tEnd[tid] = (tid < L) ? T[ENDL * L + tid] : 0.0f;

    // --- A-fragments of E = exp(T), fixed per wave: 13 chunks x v2f -------
    // A 16x4 layout: lanes 0-15 hold (K=4kt,4kt+1), lanes 16-31 (4kt+2,4kt+3)
    v2f efrag[KCH];
    {
        const int row = wid * 16 + col;        // M row of this lane
        #pragma unroll
        for (int kt = 0; kt < KCH; ++kt) {
            int j = kt * 4 + half * 2;
            float e0 = 0.0f, e1 = 0.0f;
            if (row < L) {
                if (j     < L) e0 = __expf(T[row * L + j]);
                if (j + 1 < L) e1 = __expf(T[row * L + j + 1]);
            }
            v2f e = {e0, e1};
            efrag[kt] = e;
        }
    }

    const int lenc = lens[b0 + col];

    // --- alpha init (D layout: reg r -> row 16w + r + 8*half, col) --------
    float a[8];
    #pragma unroll
    for (int r = 0; r < 8; ++r) {
        int i = wid * 16 + r + half * 8;
        a[r] = (i == START) ? 0.0f : ((i < L) ? -100.0f : -1e30f);
    }
    {   // initial per-column partial max
        float p = a[0];
        #pragma unroll
        for (int r = 1; r < 8; ++r) p = fmaxf(p, a[r]);
        pm[wid * 32 + lane] = p;
    }

    // --- preload logit tile for t=0 via TDM (wave 0 only) -----------------
    if (wid == 0)
        tdm_load_tile(logits + ((size_t)b0 * SSEQ + 0) * NL,
                      (unsigned)(uintptr_t)&lbuf[0][0]);
    __syncthreads();

    for (int t = 0; t < SSEQ; ++t) {
        // m[col] = max over all 64 rows (8 partials)
        float m = pm[col];
        #pragma unroll
        for (int k = 1; k < 8; ++k) m = fmaxf(m, pm[col + 16 * k]);

        // v = exp(alpha - m), stored in B-fragment pair layout
        #pragma unroll
        for (int r = 0; r < 8; ++r) {
            int j = wid * 16 + r + half * 8;
            vbuf[((j >> 1) * 16 + col) * 2 + (j & 1)] = __expf(a[r] - m);
        }

        // prefetch next logit tile, ensure current one has landed
        if (wid == 0) {
            if (t + 1 < SSEQ) {
                tdm_load_tile(logits + ((size_t)b0 * SSEQ + (t + 1)) * NL,
                              (unsigned)(uintptr_t)&lbuf[(t + 1) & 1][0]);
                __builtin_amdgcn_s_wait_tensorcnt(1);
            } else {
                __builtin_amdgcn_s_wait_tensorcnt(0);
            }
        }
        __syncthreads();   // vbuf ready, lbuf[t&1] ready, pm reads done

        // --- s = E · v : 13 chained f32 WMMAs (chunk 12 covers j=48..51,
        //     j=51 is the zero pad column of v; chunks 13..15 all padding) --
        v8f c = {0.f, 0.f, 0.f, 0.f, 0.f, 0.f, 0.f, 0.f};
        #pragma unroll
        for (int kt = 0; kt < KCH; ++kt) {
            int pair = kt * 2 + half;
            v2f bfrag = *(const v2f*)&vbuf[(pair * 16 + col) * 2];
            c = __builtin_amdgcn_wmma_f32_16x16x4_f32(
                    false, efrag[kt], false, bfrag, (short)0, c, false, false);
        }

        // --- alpha' = lp + m + log(s), with freeze mask -----------------
        const float* lb = &lbuf[t & 1][0];
        const bool keep = (t < lenc);
        float p = -3.0e38f;
        #pragma unroll
        for (int r = 0; r < 8; ++r) {
            int i = wid * 16 + r + half * 8;
            float lv = lb[col * NL + (i < NL ? i : NL - 1)];
            float lp = (i < NL) ? lv : -100.0f;     // START/END pad = NEG
            float an = lp + m + __logf(c[r]);
            an = (i < L) ? fmaxf(an, -1e30f) : -1e30f;  // clamp -inf/pads
            a[r] = keep ? an : a[r];
            p = fmaxf(p, a[r]);
        }
        pm[wid * 32 + lane] = p;
        __syncthreads();   // publish pm; vbuf reads done before next write
    }

    // --- norm = logsumexp(alpha + T[END][:]) ------------------------------
    float p = -3.0e38f;
    #pragma unroll
    for (int r = 0; r < 8; ++r) {
        int i = wid * 16 + r + half * 8;
        a[r] += tEnd[i];
        p = fmaxf(p, a[r]);
    }
    pm[wid * 32 + lane] = p;
    __syncthreads();

    float m = pm[col];
    #pragma unroll
    for (int k = 1; k < 8; ++k) m = fmaxf(m, pm[col + 16 * k]);
    float s = 0.0f;
    #pragma unroll
    for (int r = 0; r < 8; ++r) s += __expf(a[r] - m);
    psum[wid * 32 + lane] = s;
    __syncthreads();

    float tot = 0.0f;
    #pragma unroll
    for (int k = 0; k < 8; ++k) tot += psum[col + 16 * k];
    float norm = m + __logf(tot);

    if (wid == 0 && lane < 16)
        out[b0 + col] = gold[b0 + col] - norm;
}

// ---------------------------------------------------------------------------
extern "C" void kernel_launch(void* const* d_in, const int* in_sizes, int n_in,
                              void* d_out, int out_size, void* d_ws, size_t ws_size,
                              hipStream_t stream)
{
    const float* logits = (const float*)d_in[0];   // [1024,512,49] f32
    const float* T      = (const float*)d_in[1];   // [51,51] f32
    const int*   labels = (const int*)d_in[2];     // [1024,512] i32
    const int*   lens   = (const int*)d_in[3];     // [1024] i32
    float*       out    = (float*)d_out;           // [1024] f32
    float*       gold   = (float*)d_ws;            // 1024 f32 scratch

    crf_gold_kernel<<<BB / 8, 256, 0, stream>>>(logits, T, labels, lens, gold);
    crf_fwd_kernel<<<BB / 16, 128, 0, stream>>>(logits, T, lens, gold, out);
}